// TimeSeriesCombined_65635690217840
// MI455X (gfx1250) — compile-verified
//
#include <hip/hip_runtime.h>
#include <hip/hip_bf16.h>

typedef __attribute__((ext_vector_type(2))) float v2f;
typedef __attribute__((ext_vector_type(8))) float v8f;
typedef __attribute__((ext_vector_type(4))) unsigned int v4u;
typedef __attribute__((ext_vector_type(8))) int v8i;
typedef __attribute__((ext_vector_type(4))) int v4i;

#define B_   64
#define T_   256
#define DIN_ 128
#define H_   256
#define G3_  768
#define DZ_  8
#define DX_  8
#define NHEAD_ 96

__device__ __forceinline__ v8f wmma_f32_k4(v2f a, v2f b, v8f c) {
  // D = A(16x4) * B(4x16) + C(16x16), fp32
  return __builtin_amdgcn_wmma_f32_16x16x4_f32(
      /*neg_a=*/false, a, /*neg_b=*/false, b,
      /*c_mod=*/(short)0, c, /*reuse_a=*/false, /*reuse_b=*/false);
}

// ---------------------------------------------------------------------------
// TDM: async-load a 2D tile [16 rows x 768 fp32], row stride T*768 elems,
// from global memory into LDS. Tracked with TENSORcnt.
// ---------------------------------------------------------------------------
__device__ __forceinline__ void tdm_load_xp_tile(const float* gptr, unsigned lds_off) {
  unsigned long long ga = (unsigned long long)(uintptr_t)gptr;
  v4u g0;
  g0[0] = 1u;                                     // count=1 (valid descriptor)
  g0[1] = lds_off;                                // lds_addr (bytes)
  g0[2] = (unsigned)(ga & 0xffffffffu);           // global_addr[31:0]
  g0[3] = (unsigned)((ga >> 32) & 0x01ffffffu)    // global_addr[56:32]
          | 0x80000000u;                          // type = 2 ("image")
  v8i g1;
  g1[0] = (int)(2u << 16);                        // data_size = 2 -> 4 bytes
  g1[1] = (int)((unsigned)G3_ << 16);             // tensor_dim0 = 768 (lo16)
  g1[2] = (int)(16u << 16);                       // tensor_dim0 hi | tensor_dim1 = 16
  g1[3] = (int)((unsigned)G3_ << 16);             // tensor_dim1 hi | tile_dim0 = 768
  g1[4] = 16;                                     // tile_dim1 = 16, tile_dim2 = 0
  g1[5] = (int)(T_ * G3_);                        // tensor_dim0_stride lo32 = 196608
  g1[6] = 0;                                      // stride0 hi | stride1 lo
  g1[7] = 0;                                      // stride1 hi
  v4i g2 = {0, 0, 0, 0};
  v4i g3 = {0, 0, 0, 0};
#if __clang_major__ >= 23
  v8i gx = {0, 0, 0, 0, 0, 0, 0, 0};
  __builtin_amdgcn_tensor_load_to_lds(g0, g1, g2, g3, gx, 0);
#else
  __builtin_amdgcn_tensor_load_to_lds(g0, g1, g2, g3, 0);
#endif
}

// ---------------------------------------------------------------------------
// Kernel 1: x_proj[M=16384, N=768] = y[M,128] @ W_ih^T + b_ih
// grid (1024, 6), block 256 (8 waves). wave -> N-tile = blockIdx.y*8 + wave.
// ---------------------------------------------------------------------------
__global__ void __launch_bounds__(256) xproj_kernel(
    const float* __restrict__ y, const float* __restrict__ Wih,
    const float* __restrict__ bih, float* __restrict__ xproj) {
  int lane = threadIdx.x & 31;
  int wave = threadIdx.x >> 5;
  int mt = blockIdx.x;                 // 0..1023
  int nt = blockIdx.y * 8 + wave;      // 0..47
  int m16 = lane & 15;
  int hi  = lane >> 4;
  int n = nt * 16 + m16;

  const float* arow = y   + (size_t)(mt * 16 + m16) * DIN_ + 2 * hi;
  const float* brow = Wih + (size_t)n * DIN_ + 2 * hi;

  v8f acc = {};
#pragma unroll 8
  for (int k = 0; k < DIN_; k += 4) {
    v2f a, b;
    a.x = arow[k]; a.y = arow[k + 1];
    b.x = brow[k]; b.y = brow[k + 1];
    acc = wmma_f32_k4(a, b, acc);
  }
  float bias = bih[n];
#pragma unroll
  for (int r = 0; r < 8; ++r) {
    int m = mt * 16 + r + 8 * hi;
    xproj[(size_t)m * G3_ + n] = acc[r] + bias;
  }
}

// ---------------------------------------------------------------------------
// Kernel 2: GRU recurrence. 4 blocks x 256 threads; block handles 16 batch
// rows. h[16,256] and pre[16,768] in static LDS; x_proj step tiles are
// double-buffered in dynamic LDS via TDM (tensor_load_to_lds), prefetching
// step t+1 while WMMA computes step t.
// ---------------------------------------------------------------------------
__global__ void __launch_bounds__(256) gru_kernel(
    const float* __restrict__ xproj, const float* __restrict__ Whh,
    const float* __restrict__ bhh, float* __restrict__ enc) {
  __shared__ float h_lds[16 * H_];     // 16 KB
  __shared__ float pre[16 * G3_];      // 48 KB
  extern __shared__ float xbuf[];      // 2 * 16*768 floats = 96 KB (dynamic)

  int tid  = threadIdx.x;
  int lane = tid & 31;
  int wave = tid >> 5;
  int m16  = lane & 15;
  int hi   = lane >> 4;
  int bbase = blockIdx.x * 16;

  for (int i = tid; i < 16 * H_; i += 256) h_lds[i] = 0.0f;

  // LDS byte offsets of the two tile buffers (flat LDS addr truncates to offset)
  unsigned xb_off0 = (unsigned)(uintptr_t)(&xbuf[0]);
  unsigned xb_off1 = xb_off0 + 16u * G3_ * sizeof(float);

  // Prefetch tile for t = 0 (rows (bbase+m)*T + 0)
  if (wave == 0) {
    tdm_load_xp_tile(xproj + (size_t)bbase * T_ * G3_, xb_off0);
  }
  __syncthreads();

  for (int t = 0; t < T_; ++t) {
    // --- issue async prefetch of x_proj tile for step t+1 (double buffer)
    if (wave == 0 && (t + 1) < T_) {
      unsigned dst = ((t + 1) & 1) ? xb_off1 : xb_off0;
      tdm_load_xp_tile(xproj + ((size_t)bbase * T_ + (t + 1)) * G3_, dst);
    }

    // --- matrix phase: 48 N-tiles split over 8 waves (uniform control flow)
    for (int nt = wave; nt < 48; nt += 8) {
      int n = nt * 16 + m16;
      const float* brow = Whh + (size_t)n * H_ + 2 * hi;
      const float* arow = h_lds + m16 * H_ + 2 * hi;
      v8f acc = {};
#pragma unroll 8
      for (int k = 0; k < H_; k += 4) {
        v2f a, b;
        a.x = arow[k]; a.y = arow[k + 1];
        b.x = brow[k]; b.y = brow[k + 1];
        acc = wmma_f32_k4(a, b, acc);
      }
      float bias = bhh[n];
#pragma unroll
      for (int r = 0; r < 8; ++r) {
        pre[(r + 8 * hi) * G3_ + n] = acc[r] + bias;
      }
    }

    // --- wave 0 ensures this step's tile landed (allow 1 in-flight prefetch)
    if (wave == 0) {
      if ((t + 1) < T_) __builtin_amdgcn_s_wait_tensorcnt(1);
      else              __builtin_amdgcn_s_wait_tensorcnt(0);
    }
    __syncthreads();

    // --- gate phase: 16*256 = 4096 states, 16 per thread; xp from LDS
    const float* xpb = xbuf + (size_t)(t & 1) * (16 * G3_);
    for (int idx = tid; idx < 16 * H_; idx += 256) {
      int m = idx >> 8;
      int j = idx & (H_ - 1);
      const float* xp = xpb + m * G3_;
      float xr = xp[j], xz = xp[j + H_], xn = xp[j + 2 * H_];
      float hr = pre[m * G3_ + j];
      float hz = pre[m * G3_ + j + H_];
      float hn = pre[m * G3_ + j + 2 * H_];
      float rg = 1.0f / (1.0f + __expf(-(xr + hr)));
      float zg = 1.0f / (1.0f + __expf(-(xz + hz)));
      float ng = tanhf(xn + rg * hn);
      float hprev = h_lds[idx];
      float hnew = (1.0f - zg) * ng + zg * hprev;
      h_lds[idx] = hnew;
      enc[((size_t)(bbase + m) * T_ + t) * H_ + j] = hnew;
    }
    __syncthreads();
  }
}

// ---------------------------------------------------------------------------
// Kernel 3: heads[16384, 96] = enc[16384,256] @ concat(Wmz,Wmx,Wpz,Wcx)^T + b
// grid 1024, block 192 (6 waves -> 6 N-tiles of 16).
// ---------------------------------------------------------------------------
__global__ void __launch_bounds__(192) heads_kernel(
    const float* __restrict__ enc,
    const float* __restrict__ Wmz, const float* __restrict__ bmz,
    const float* __restrict__ Wmx, const float* __restrict__ bmx,
    const float* __restrict__ Wpz, const float* __restrict__ bpz,
    const float* __restrict__ Wcx, const float* __restrict__ bcx,
    float* __restrict__ hw) {
  int lane = threadIdx.x & 31;
  int wave = threadIdx.x >> 5;
  int m16 = lane & 15;
  int hi  = lane >> 4;
  int mt = blockIdx.x;
  int n = wave * 16 + m16;             // 0..95

  const float* wrow;
  float bias;
  if (n < 8)        { wrow = Wmz + (size_t)n * H_;        bias = bmz[n]; }
  else if (n < 16)  { wrow = Wmx + (size_t)(n - 8) * H_;  bias = bmx[n - 8]; }
  else if (n < 32)  { wrow = Wpz + (size_t)(n - 16) * H_; bias = bpz[n - 16]; }
  else              { wrow = Wcx + (size_t)(n - 32) * H_; bias = bcx[n - 32]; }
  wrow += 2 * hi;

  const float* arow = enc + (size_t)(mt * 16 + m16) * H_ + 2 * hi;
  v8f acc = {};
#pragma unroll 8
  for (int k = 0; k < H_; k += 4) {
    v2f a, b;
    a.x = arow[k]; a.y = arow[k + 1];
    b.x = wrow[k]; b.y = wrow[k + 1];
    acc = wmma_f32_k4(a, b, acc);
  }
#pragma unroll
  for (int r = 0; r < 8; ++r) {
    int row = mt * 16 + r + 8 * hi;
    hw[(size_t)row * NHEAD_ + n] = acc[r] + bias;
  }
}

// ---------------------------------------------------------------------------
// Kernel 4: chol_z [DZ,B,T,T] — streaming float4 fill (134 MB, HBM-bound).
// diag[d,b,t,t] = softplus(bd_z[b,t,d]);  off[d,b,t,t+1] = bd_z[b,t,DZ+d].
// ---------------------------------------------------------------------------
__global__ void __launch_bounds__(256) chol_kernel(
    const float* __restrict__ hw, float* __restrict__ chol) {
  size_t idx = (size_t)blockIdx.x * blockDim.x + threadIdx.x; // 8,388,608
  size_t row = idx >> 6;               // (d*B + b)*T + t
  int c4 = (int)(idx & 63);
  int t = (int)(row & (T_ - 1));
  size_t db = row >> 8;
  int b = (int)(db & (B_ - 1));
  int d = (int)(db >> 6);
  int base = c4 * 4;

  float v0 = 0.f, v1 = 0.f, v2 = 0.f, v3 = 0.f;
  if (t >= base - 1 && t <= base + 3) {
    const float* hp = hw + (size_t)(b * T_ + t) * NHEAD_;
    float dg = hp[16 + d];
    dg = (dg > 20.0f) ? dg : log1pf(__expf(dg));
    float of = (t < T_ - 1) ? hp[24 + d] : 0.0f;
    float vals[4] = {0.f, 0.f, 0.f, 0.f};
    int dcol = t - base;
    int ocol = t + 1 - base;
    if (dcol >= 0 && dcol < 4) vals[dcol] = dg;
    if (ocol >= 0 && ocol < 4) vals[ocol] = of;
    v0 = vals[0]; v1 = vals[1]; v2 = vals[2]; v3 = vals[3];
  }
  *(float4*)(chol + (idx << 2)) = make_float4(v0, v1, v2, v3);
}

// ---------------------------------------------------------------------------
// Kernel 5: mean_z / anchored mean_x / cov_x from head workspace.
// ---------------------------------------------------------------------------
__global__ void __launch_bounds__(256) assemble_kernel(
    const float* __restrict__ hw, float* __restrict__ meanz,
    float* __restrict__ meanx, float* __restrict__ covx) {
  int idx = blockIdx.x * blockDim.x + threadIdx.x;
  if (idx >= 16384 * 80) return;
  int bt = idx / 80;
  int c  = idx % 80;
  int b  = bt >> 8;
  if (c < 8) {
    meanz[bt * 8 + c] = hw[(size_t)bt * NHEAD_ + c];
  } else if (c < 16) {
    int cc = c - 8;
    float v = hw[(size_t)bt * NHEAD_ + 8 + cc];
    if (cc < 2) v -= hw[(size_t)(b * T_) * NHEAD_ + 8 + cc];
    meanx[bt * 8 + cc] = v;
  } else {
    int j = c - 16;
    covx[(size_t)bt * 64 + j] = hw[(size_t)bt * NHEAD_ + 32 + j];
  }
}

extern "C" void kernel_launch(void* const* d_in, const int* in_sizes, int n_in,
                              void* d_out, int out_size, void* d_ws, size_t ws_size,
                              hipStream_t stream) {
  const float* y    = (const float*)d_in[0];
  const float* Wih  = (const float*)d_in[1];
  const float* Whh  = (const float*)d_in[2];
  const float* bih  = (const float*)d_in[3];
  const float* bhh  = (const float*)d_in[4];
  const float* Wmz  = (const float*)d_in[5];
  const float* bmz  = (const float*)d_in[6];
  const float* Wmx  = (const float*)d_in[7];
  const float* bmx  = (const float*)d_in[8];
  const float* Wpz  = (const float*)d_in[9];
  const float* bpz  = (const float*)d_in[10];
  const float* Wcx  = (const float*)d_in[11];
  const float* bcx  = (const float*)d_in[12];

  float* ws    = (float*)d_ws;
  float* xproj = ws;                                   // 12,582,912 f
  float* enc   = ws + (size_t)12582912;                //  4,194,304 f
  float* hw    = ws + (size_t)12582912 + 4194304;      //  1,572,864 f

  float* out   = (float*)d_out;
  float* meanz = out;                                  //    131,072 f
  float* chol  = out + (size_t)131072;                 // 33,554,432 f
  float* meanx = out + (size_t)131072 + 33554432;      //    131,072 f
  float* covx  = out + (size_t)131072 + 33554432 + 131072; // 1,048,576 f

  xproj_kernel<<<dim3(1024, 6), 256, 0, stream>>>(y, Wih, bih, xproj);
  gru_kernel<<<4, 256, 2 * 16 * G3_ * sizeof(float), stream>>>(xproj, Whh, bhh, enc);
  heads_kernel<<<1024, 192, 0, stream>>>(enc, Wmz, bmz, Wmx, bmx,
                                         Wpz, bpz, Wcx, bcx, hw);
  chol_kernel<<<32768, 256, 0, stream>>>(hw, chol);
  assemble_kernel<<<(16384 * 80 + 255) / 256, 256, 0, stream>>>(hw, meanz, meanx, covx);
}